// UpsampleFlow_71863392797318
// MI455X (gfx1250) — compile-verified
//
#include <hip/hip_runtime.h>

typedef __attribute__((ext_vector_type(2))) float v2f;
typedef __attribute__((ext_vector_type(8))) float v8f;

#define EPSF 1e-10f
#define IDX_MASK 0xFFFu          // S <= 4096 sparse indices in low 12 bits
#define KEY_MASK 0xFFFFF000u

// ---------------------------------------------------------------------------
// Prep: pack sparse points as A-matrix rows (-2x, -2y, -2z, |s|^2)
// so D = A*B + C gives  |x|^2 - 2 x.s + |s|^2 + 1  (true sqdist, +1 bias so
// f32 bits of every candidate compare monotonically as unsigned).
// ---------------------------------------------------------------------------
__global__ void uf_pack_sparse(const float* __restrict__ sxyz,
                               float* __restrict__ wsS, int B, int S) {
    int t = blockIdx.x * blockDim.x + threadIdx.x;
    if (t >= B * S) return;
    int b = t / S, s = t - b * S;
    const float* base = sxyz + (size_t)b * 3 * S;
    float x = base[s], y = base[S + s], z = base[2 * S + s];
    float4 o = make_float4(-2.0f * x, -2.0f * y, -2.0f * z, x * x + y * y + z * z);
    *(float4*)(wsS + ((size_t)t << 2)) = o;
}

// Prep: pack dense points as B-matrix columns (x, y, z, 1)
__global__ void uf_pack_dense(const float* __restrict__ xyz,
                              float* __restrict__ wsD, int B, int N) {
    int t = blockIdx.x * blockDim.x + threadIdx.x;
    if (t >= B * N) return;
    int b = t / N, n = t - b * N;
    const float* base = xyz + (size_t)b * 3 * N;
    float4 o = make_float4(base[n], base[N + n], base[2 * N + n], 1.0f);
    *(float4*)(wsD + ((size_t)t << 2)) = o;
}

static __device__ __forceinline__ unsigned umn(unsigned a, unsigned b) { return a < b ? a : b; }
static __device__ __forceinline__ unsigned umx(unsigned a, unsigned b) { return a > b ? a : b; }
// canonical med3 pattern -> backend folds to v_med3_u32
static __device__ __forceinline__ unsigned umed3(unsigned a, unsigned b, unsigned c) {
    return umx(umn(a, b), umn(umx(a, b), c));
}
// insert key k into sorted running-min triple (m0 <= m1 <= m2), branchless
static __device__ __forceinline__ void ins3(unsigned k, unsigned& m0, unsigned& m1, unsigned& m2) {
    unsigned t2 = umed3(m1, m2, k);   // uses old m1, m2
    unsigned t1 = umed3(m0, m1, k);   // uses old m0, m1
    unsigned t0 = umn(m0, k);
    m0 = t0; m1 = t1; m2 = t2;
}

// ---------------------------------------------------------------------------
// Main: one wave handles 16 dense points; loops over S in 16-sparse chunks.
// WMMA D-tile layout (32b C/D, wave32): lane holds column n = lane&15,
// rows M = vgpr + 8*(lane>=16).  -> lane-local branchless top-3 over packed
// (distance|index) keys, then one xor-16 shuffle merge with the partner lane.
// ---------------------------------------------------------------------------
__global__ void __launch_bounds__(128)
uf_knn3_interp(const float* __restrict__ wsS, const float* __restrict__ wsD,
               const float* __restrict__ sxyz, const float* __restrict__ sflow,
               float* __restrict__ out, int B, int N, int S, int totalTiles) {
    const int lane = threadIdx.x & 31;
    const int wave = threadIdx.x >> 5;
    const int tile = blockIdx.x * 4 + wave;
    if (tile >= totalTiles) return;               // whole-wave exit: EXEC all-1 for WMMA

    const int tilesPerB = N >> 4;
    const int b  = tile / tilesPerB;
    const int n0 = (tile - b * tilesPerB) << 4;
    const int half = lane >> 4;                   // 0: K=0,1 / rows 0..7 ; 1: K=2,3 / rows 8..15
    const int l16  = lane & 15;
    const int n    = n0 + l16;

    // Dense point for this lane's column (packed x,y,z,1)
    float4 dp = *(const float4*)(wsD + (((size_t)b * N + n) << 2));
    const float px = dp.x, py = dp.y, pz = dp.z;

    // B fragment (loop invariant): lane -> col n, comps K=2*half..2*half+1
    v2f bfrag;
    bfrag.x = half ? pz : px;
    bfrag.y = half ? 1.0f : py;

    // C accumulator: broadcast |x|^2 + 1 -> D = sqdist + 1 >= ~1 (positive)
    const float cbias = px * px + py * py + pz * pz + 1.0f;
    v8f c8;
#pragma unroll
    for (int v = 0; v < 8; ++v) c8[v] = cbias;

    unsigned m0 = 0xFFFFFFFFu, m1 = 0xFFFFFFFFu, m2 = 0xFFFFFFFFu;
    const int rowOff = half << 3;
    const float* aBase = wsS + (((size_t)b * S) << 2) + (l16 << 2) + 2 * half;

#pragma unroll 2
    for (int s0 = 0; s0 < S; s0 += 16) {
        // A fragment (sparse rows s0..s0+15): lane -> row s0+l16, comps K=2*half..+1
        v2f afrag = *(const v2f*)(aBase + ((size_t)s0 << 2));
        v8f acc = __builtin_amdgcn_wmma_f32_16x16x4_f32(false, afrag, false, bfrag,
                                                        (short)0, c8, false, false);
        const unsigned sbase = (unsigned)(s0 + rowOff);   // 8-aligned -> OR-exact with v
#pragma unroll
        for (int v = 0; v < 8; ++v) {
            unsigned k = (__float_as_uint(acc[v]) & KEY_MASK) | sbase | (unsigned)v;
            ins3(k, m0, m1, m2);
        }
    }

    // Merge the partner lane's top-3 (snapshot before mutating).
    unsigned p0 = (unsigned)__shfl_xor((int)m0, 16, 32);
    unsigned p1 = (unsigned)__shfl_xor((int)m1, 16, 32);
    unsigned p2 = (unsigned)__shfl_xor((int)m2, 16, 32);
    ins3(p0, m0, m1, m2);
    ins3(p1, m0, m1, m2);
    ins3(p2, m0, m1, m2);

    // Finalize: exact distances for the 3 winners, inverse-distance weights.
    const float* sxb = sxyz + (size_t)b * 3 * S;
    const float* sfb = sflow + (size_t)b * 3 * S;

    int idx[3] = { (int)(m0 & IDX_MASK), (int)(m1 & IDX_MASK), (int)(m2 & IDX_MASK) };
    float inv[3];
    float wsum = 0.0f;
#pragma unroll
    for (int j = 0; j < 3; ++j) {
        int s = idx[j];
        float ax = sxb[s]         - px;
        float ay = sxb[S + s]     - py;
        float az = sxb[2 * S + s] - pz;
        float dd = sqrtf(ax * ax + ay * ay + az * az);
        dd = fmaxf(dd, EPSF);
        float iv = 1.0f / dd;
        inv[j] = iv;
        wsum += iv;
    }
    float r = 1.0f / wsum;
    float fx = 0.0f, fy = 0.0f, fz = 0.0f;
#pragma unroll
    for (int j = 0; j < 3; ++j) {
        int s = idx[j];
        float wj = inv[j] * r;
        fx += wj * sfb[s];
        fy += wj * sfb[S + s];
        fz += wj * sfb[2 * S + s];
    }
    if (half == 0) {
        float* ob = out + (size_t)b * 3 * N;
        ob[n]         = fx;
        ob[N + n]     = fy;
        ob[2 * N + n] = fz;
    }
}

extern "C" void kernel_launch(void* const* d_in, const int* in_sizes, int n_in,
                              void* d_out, int out_size, void* d_ws, size_t ws_size,
                              hipStream_t stream) {
    const float* xyz   = (const float*)d_in[0];
    const float* sxyz  = (const float*)d_in[1];
    const float* sflow = (const float*)d_in[2];
    float* out = (float*)d_out;

    const int B = 2, C = 3;
    const int N = in_sizes[0] / (B * C);   // 16384
    const int S = in_sizes[1] / (B * C);   // 4096

    float* wsS = (float*)d_ws;                      // B*S*4 floats
    float* wsD = wsS + (size_t)B * S * 4;           // B*N*4 floats

    uf_pack_sparse<<<(B * S + 255) / 256, 256, 0, stream>>>(sxyz, wsS, B, S);
    uf_pack_dense <<<(B * N + 255) / 256, 256, 0, stream>>>(xyz, wsD, B, N);

    const int totalTiles = B * (N >> 4);            // 2048 waves of work
    const int blocks = (totalTiles + 3) / 4;        // 4 waves / 128-thread block
    uf_knn3_interp<<<blocks, 128, 0, stream>>>(wsS, wsD, sxyz, sflow, out,
                                               B, N, S, totalTiles);
}